// LoRA_76072460747085
// MI455X (gfx1250) — compile-verified
//
#include <hip/hip_runtime.h>

// ---------------------------------------------------------------------------
// LoRA fused GEMM for MI455X (gfx1250, wave32, WMMA + TDM):
//   out[M,N] = x[M,K] @ (W[N,K] + A[N,R] @ B[R,K])^T
//            = x @ W^T  +  (x @ B^T) @ A^T
// M=8192, N=4096, K=4096, R=16.  HBM-bound (~320MB @ 23.3TB/s => ~14us),
// so full f32 precision via V_WMMA_F32_16X16X4_F32; W_eff never materialized.
// Global->LDS staging is done by the Tensor Data Mover (tensor_load_to_lds,
// 6-arg clang-23 form), double-buffered and overlapped with WMMA compute,
// tracked via TENSORcnt.
// ---------------------------------------------------------------------------

typedef float        v2f __attribute__((ext_vector_type(2)));
typedef float        v8f __attribute__((ext_vector_type(8)));
typedef unsigned int v4u __attribute__((ext_vector_type(4)));
typedef int          v4i __attribute__((ext_vector_type(4)));
typedef int          v8i __attribute__((ext_vector_type(8)));

#define WMMA_F32(A_, B_, C_) \
    __builtin_amdgcn_wmma_f32_16x16x4_f32(false, (A_), false, (B_), (short)0, (C_), false, false)

namespace {
constexpr int Mtok = 8192;   // n_tok
constexpr int Nout = 4096;   // d_out
constexpr int Kin  = 4096;   // d_in
constexpr int R    = 16;     // LoRA rank

constexpr int BM = 128;      // block tile M
constexpr int BN = 128;      // block tile N
constexpr int BK = 32;       // K slab per pipeline stage

// LDS pitches (floats).  34 = 32 data + 2 pad DWORDs, produced by the TDM's
// pad_interval=32-DWORD / pad_amount=2-DWORD feature; 16 strided rows then
// hit 16 distinct banks for the ds_load fragment reads.
constexpr int XP = 34;
constexpr int TP = 18;       // pitch for t = x@B^T staging tile
constexpr int NSLAB = Kin / BK;   // 128
}

// Issue one 2-D TDM tile load: global (row-major, rowStride elems) -> LDS with
// 2-DWORD padding after every 32 DWORDs (so LDS pitch = tileW + 2 floats).
// D# layout per CDNA5 ISA 8.3/8.4; groups 2/3 zero (2-D tensor), type=2.
__device__ __forceinline__ void tdm_load_tile_f32(
    unsigned int lds_byte_addr,           // LDS dest offset (bytes)
    const float* tile_start,              // global address of tile start
    unsigned int tileW,                   // tile dim0 (elements) == 32 here
    unsigned int tileH,                   // tile dim1 (rows)
    unsigned int tensorW,                 // tensor dim0 (elements)
    unsigned int tensorH,                 // tensor dim1 (rows)
    unsigned long long rowStride)         // tensor dim0 stride (elements)
{
    const unsigned long long ga = (unsigned long long)(uintptr_t)tile_start;

    v4u g0;
    g0[0] = 1u;                                            // count=1 (valid), user mode
    g0[1] = lds_byte_addr;                                 // lds_addr
    g0[2] = (unsigned int)(ga & 0xFFFFFFFFu);              // global_addr[31:0]
    g0[3] = (unsigned int)((ga >> 32) & 0x01FFFFFFu)       // global_addr[56:32]
          | (2u << 30);                                    // type=2 ("image")

    v8i g1;
    g1[0] = (int)((2u << 16)        // data_size = 4 bytes
                | (1u << 20)        // pad_enable
                | (4u << 22)        // pad_interval code 4 -> every 32 DWORDs
                | (1u << 25));      // pad_amount  code 1 -> 2 DWORDs
    g1[1] = (int)((tensorW & 0xFFFFu) << 16);              // abar=0 | tensor_dim0[15:0]
    g1[2] = (int)(((tensorW >> 16) & 0xFFFFu) | ((tensorH & 0xFFFFu) << 16));
    g1[3] = (int)(((tensorH >> 16) & 0xFFFFu) | (tileW << 16));   // | tile_dim0
    g1[4] = (int)(tileH & 0xFFFFu);                        // tile_dim1 | tile_dim2=0
    g1[5] = (int)(rowStride & 0xFFFFFFFFull);              // dim0_stride[31:0]
    g1[6] = (int)((rowStride >> 32) & 0xFFFFull);          // dim0_stride[47:32] | dim1_stride=0
    g1[7] = 0;

    const v4i gz4 = {0, 0, 0, 0};                          // groups 2/3: unused (2-D)
    const v8i gz8 = {0, 0, 0, 0, 0, 0, 0, 0};              // trailing group (unused)
    __builtin_amdgcn_tensor_load_to_lds(g0, g1, gz4, gz4, gz8, 0 /*cpol*/);
}

__global__ __launch_bounds__(256)
void lora_fused_gemm_f32(const float* __restrict__ xg,   // [M,K]
                         const float* __restrict__ wg,   // [N,K]
                         const float* __restrict__ ag,   // [N,R]
                         const float* __restrict__ bg,   // [R,K]
                         float* __restrict__ outg)       // [M,N]
{
    __shared__ float xsh[2][BM * XP];    // 2 x 17408 B  (TDM-written, pitch 34)
    __shared__ float wsh[2][BN * XP];    // 2 x 17408 B
    __shared__ float bsh[2][R  * XP];    // 2 x  2176 B
    __shared__ float tsh[BM * TP];       //      9216 B  (t = x_tile @ B^T)

    const int tid    = threadIdx.x;        // 0..255 (8 wave32s)
    const int lane   = tid & 31;
    const int wave   = tid >> 5;           // 0..7
    const int wave_m = wave >> 1;          // 0..3  -> 32 rows each
    const int wave_n = wave & 1;           // 0..1  -> 64 cols each
    const int l15    = lane & 15;
    const int lh     = lane >> 4;          // 0/1 half-wave

    const int bm  = blockIdx.y * BM;       // token-row base
    const int bn  = blockIdx.x * BN;       // d_out-col base
    const int wm0 = wave_m * 32;
    const int wn0 = wave_n * 64;

    v8f acc[2][4];                         // 2x4 tiles of 16x16 per wave
    v8f tacc[2];                           // x@B^T accumulators (N-dim = r)
#pragma unroll
    for (int i = 0; i < 2; ++i) {
        tacc[i] = (v8f)0.0f;
#pragma unroll
        for (int j = 0; j < 4; ++j) acc[i][j] = (v8f)0.0f;
    }

    // LDS byte offsets for the TDM descriptors (wave-uniform scalars).
    const unsigned xs_lds = (unsigned)(uintptr_t)&xsh[0][0];
    const unsigned ws_lds = (unsigned)(uintptr_t)&wsh[0][0];
    const unsigned bs_lds = (unsigned)(uintptr_t)&bsh[0][0];
    constexpr unsigned XB = BM * XP * 4;   // bytes per x/w buffer
    constexpr unsigned BB = R  * XP * 4;   // bytes per lora-B buffer

    // ---------------- TDM-fed, double-buffered K pipeline -----------------
    // Wave 0 DMAs slab s+1 while everyone computes slab s.  TDM ops complete
    // in order per wave, so s_wait_tensorcnt(3) retires exactly slab s.
    if (wave == 0) {
        tdm_load_tile_f32(xs_lds, xg + (size_t)bm * Kin, BK, BM, Kin, Mtok, Kin);
        tdm_load_tile_f32(ws_lds, wg + (size_t)bn * Kin, BK, BN, Kin, Nout, Kin);
        tdm_load_tile_f32(bs_lds, bg,                    BK, R,  Kin, R,    Kin);
    }

    for (int s = 0; s < NSLAB; ++s) {
        const int buf = s & 1;

        if (wave == 0) {
            if (s + 1 < NSLAB) {
                const int k1 = (s + 1) * BK;
                const unsigned nb = (unsigned)(buf ^ 1);
                tdm_load_tile_f32(xs_lds + nb * XB, xg + (size_t)bm * Kin + k1,
                                  BK, BM, Kin, Mtok, Kin);
                tdm_load_tile_f32(ws_lds + nb * XB, wg + (size_t)bn * Kin + k1,
                                  BK, BN, Kin, Nout, Kin);
                tdm_load_tile_f32(bs_lds + nb * BB, bg + k1,
                                  BK, R,  Kin, R,    Kin);
                __builtin_amdgcn_s_wait_tensorcnt(3);  // slab s landed
            } else {
                __builtin_amdgcn_s_wait_tensorcnt(0);  // final slab landed
            }
        }
        __syncthreads();   // publish slab s

        const float* xs = &xsh[buf][0];
        const float* ws = &wsh[buf][0];
        const float* bs = &bsh[buf][0];

        // 8 k-steps of 4: 2 A-frags, 4 B-frags, 1 LoRA-frag -> 10 WMMAs.
        // Fragment = one float2 LDS read:
        //   lane L -> (row = base + (L&15), col = k + 2*(L>>4))  [ISA layout]
#pragma unroll
        for (int kk = 0; kk < BK; kk += 4) {
            const int kcol = kk + 2 * lh;
            v2f afrag[2], bfrag[4], lfrag;
            afrag[0] = *(const v2f*)&xs[(wm0      + l15) * XP + kcol];
            afrag[1] = *(const v2f*)&xs[(wm0 + 16 + l15) * XP + kcol];
            lfrag    = *(const v2f*)&bs[ l15             * XP + kcol];
#pragma unroll
            for (int j = 0; j < 4; ++j)
                bfrag[j] = *(const v2f*)&ws[(wn0 + j * 16 + l15) * XP + kcol];

#pragma unroll
            for (int i = 0; i < 2; ++i) {
                tacc[i] = WMMA_F32(afrag[i], lfrag, tacc[i]);
#pragma unroll
                for (int j = 0; j < 4; ++j)
                    acc[i][j] = WMMA_F32(afrag[i], bfrag[j], acc[i][j]);
            }
        }
        __syncthreads();   // all reads of buf done before TDM overwrites it
    }

    // --------------------- LoRA epilogue: out += t @ A^T ------------------
    // Park t (C/D layout) in LDS so it can be re-read in A-operand layout.
    // wave_n==0/1 computed identical t; one copy stores (wave-uniform branch).
    if (wave_n == 0) {
#pragma unroll
        for (int i = 0; i < 2; ++i) {
            const int mbase = wm0 + i * 16 + lh * 8;
#pragma unroll
            for (int e = 0; e < 8; ++e)
                tsh[(mbase + e) * TP + l15] = tacc[i][e];
        }
    }
    __syncthreads();

#pragma unroll
    for (int r0 = 0; r0 < R; r0 += 4) {
        const int rc = r0 + 2 * lh;
        v2f tfrag[2], afr[4];
        tfrag[0] = *(const v2f*)&tsh[(wm0      + l15) * TP + rc];
        tfrag[1] = *(const v2f*)&tsh[(wm0 + 16 + l15) * TP + rc];
#pragma unroll
        for (int j = 0; j < 4; ++j)   // LoRA-A straight from global (L2-hot, 256KB)
            afr[j] = *(const v2f*)(ag + (size_t)(bn + wn0 + j * 16 + l15) * R + rc);
#pragma unroll
        for (int i = 0; i < 2; ++i)
#pragma unroll
            for (int j = 0; j < 4; ++j)
                acc[i][j] = WMMA_F32(tfrag[i], afr[j], acc[i][j]);
    }

    // ------------------------------ store ---------------------------------
    // C/D layout: vgpr e, lane L -> row = m0 + e + 8*(L>>4), col = n0 + (L&15).
#pragma unroll
    for (int i = 0; i < 2; ++i) {
        const int m0 = bm + wm0 + i * 16 + lh * 8;
#pragma unroll
        for (int j = 0; j < 4; ++j) {
            const int n0 = bn + wn0 + j * 16 + l15;
#pragma unroll
            for (int e = 0; e < 8; ++e)
                outg[(size_t)(m0 + e) * Nout + n0] = acc[i][j][e];
        }
    }
}

extern "C" void kernel_launch(void* const* d_in, const int* in_sizes, int n_in,
                              void* d_out, int out_size, void* d_ws, size_t ws_size,
                              hipStream_t stream) {
    (void)in_sizes; (void)n_in; (void)out_size; (void)d_ws; (void)ws_size;
    const float* x = (const float*)d_in[0];   // [8192,4096]
    const float* W = (const float*)d_in[1];   // [4096,4096]
    const float* A = (const float*)d_in[2];   // [4096,16]
    const float* B = (const float*)d_in[3];   // [16,4096]
    float* out = (float*)d_out;               // [8192,4096] f32

    dim3 grid(Nout / BN, Mtok / BM);          // (32, 64)
    lora_fused_gemm_f32<<<grid, 256, 0, stream>>>(x, W, A, B, out);
}